// TruncatedDiscriminator_28664611733747
// MI455X (gfx1250) — compile-verified
//
#include <hip/hip_runtime.h>
#include <hip/hip_bf16.h>

// ---------------------------------------------------------------------------
// Truncated-signature MMD, algebraically collapsed:
//   out = || sum_b sig(mu_b) - sum_b sig(nu_b) ||^2 / B^2
// Levels (d=5, order 4): P = 5+25+125+625 = 780 (scalar term cancels).
// Phase 1: per-wave Chen scan in LDS (fp32).
// Phase 2: batch reduction via V_WMMA_F32_16X16X4_F32 with A = ones,
//          4 independent accumulators for ILP, branch-free clamped loads.
// Phase 3: deterministic sum of squares.
// ---------------------------------------------------------------------------

#define BATCH_N   2048
#define T_STREAM  128
#define CH        5
#define NLEV      780     // 5 + 25 + 125 + 625
#define O2        5
#define O3        30
#define O4        155
#define WAVES     8

typedef float v2f __attribute__((ext_vector_type(2)));
typedef float v8f __attribute__((ext_vector_type(8)));

// ---------------------------------------------------------------------------
// Chen-scan for one path (one wave32). ST = 780-float state, E = 155-float
// tensor-exponential scratch (E1|E2|E3). All cross-lane traffic via LDS;
// phase ordering enforced with __syncthreads (uniform across the block).
// ---------------------------------------------------------------------------
__device__ __forceinline__ void sig_scan(const float* __restrict__ pin,
                                         const float* sc,
                                         float* ST, float* E, int lane)
{
    // zero state
#pragma unroll
    for (int r = 0; r < 25; ++r) {
        int p = lane + 32 * r;
        if (p < NLEV) ST[p] = 0.f;
    }
    float prev[CH];
#pragma unroll
    for (int c = 0; c < CH; ++c) prev[c] = pin[c];
    __syncthreads();

    for (int t = 1; t < T_STREAM; ++t) {
        float cur[CH];
#pragma unroll
        for (int c = 0; c < CH; ++c) cur[c] = pin[t * CH + c];

        // E1 = sigma-scaled increment (lane 0 publishes; all lanes hold same data)
        if (lane == 0) {
#pragma unroll
            for (int c = 0; c < CH; ++c) E[c] = (cur[c] - prev[c]) * sc[c];
        }
#pragma unroll
        for (int c = 0; c < CH; ++c) prev[c] = cur[c];
        __syncthreads();

        // E2 = dx (x) dx / 2
        if (lane < 25) E[O2 + lane] = 0.5f * E[lane / 5] * E[lane % 5];
        __syncthreads();

        // E3 = E2 (x) dx / 3
#pragma unroll
        for (int r = 0; r < 4; ++r) {
            int p = lane + 32 * r;
            if (p < 125) E[O3 + p] = (1.f / 3.f) * E[O2 + p / 5] * E[p % 5];
        }
        __syncthreads();

        // S4 += E4 + S1(x)E3 + S2(x)E2 + S3(x)E1   (E4 = E3(x)dx/4), old S1..S3
#pragma unroll
        for (int r = 0; r < 20; ++r) {
            int p = lane + 32 * r;
            if (p < 625) {
                float e3a = E[O3 + p / 5];      // (i,j,k)
                float e3b = E[O3 + p % 125];    // (j,k,l)
                float e2v = E[O2 + p % 25];     // (k,l)
                float dxl = E[p % 5];           // l
                float s1v = ST[p / 125];
                float s2v = ST[O2 + p / 25];
                float s3v = ST[O3 + p / 5];
                ST[O4 + p] += 0.25f * e3a * dxl + s1v * e3b + s2v * e2v + s3v * dxl;
            }
        }
        __syncthreads();

        // S3 += E3 + S1(x)E2 + S2(x)E1  (old S1,S2)
#pragma unroll
        for (int r = 0; r < 4; ++r) {
            int p = lane + 32 * r;
            if (p < 125) {
                ST[O3 + p] += E[O3 + p]
                            + ST[p / 25] * E[O2 + p % 25]
                            + ST[O2 + p / 5] * E[p % 5];
            }
        }
        __syncthreads();

        // S2 += E2 + S1(x)E1  (old S1)
        if (lane < 25) ST[O2 + lane] += E[O2 + lane] + ST[lane / 5] * E[lane % 5];
        __syncthreads();

        // S1 += E1
        if (lane < CH) ST[lane] += E[lane];
        __syncthreads();
    }
}

// ---------------------------------------------------------------------------
// Phase 1: one wave per batch element; compute sig(mu) and sig(nu), store diff
// ---------------------------------------------------------------------------
__global__ __launch_bounds__(WAVES * 32)
void sigdiff_kernel(const float* __restrict__ x, const float* __restrict__ y,
                    const float* __restrict__ sigma, float* __restrict__ diff)
{
    __shared__ float STs[WAVES][800];
    __shared__ float Es[WAVES][160];

    int wave = threadIdx.x >> 5;
    int lane = threadIdx.x & 31;
    int b = blockIdx.x * WAVES + wave;

    float* ST = STs[wave];
    float* E  = Es[wave];

    float scx[CH];
    scx[0] = 1.f;
#pragma unroll
    for (int c = 1; c < CH; ++c) scx[c] = sigma[c - 1];
    float sc1[CH] = {1.f, 1.f, 1.f, 1.f, 1.f};

    // signature of mu = [x_time, x_rest * sigma]
    sig_scan(x + (size_t)b * (T_STREAM * CH), scx, ST, E, lane);

    float sv[25];
#pragma unroll
    for (int r = 0; r < 25; ++r) {
        int p = lane + 32 * r;
        sv[r] = (p < NLEV) ? ST[p] : 0.f;
    }
    __syncthreads();

    // signature of nu = y
    sig_scan(y + (size_t)b * (T_STREAM * CH), sc1, ST, E, lane);

#pragma unroll
    for (int r = 0; r < 25; ++r) {
        int p = lane + 32 * r;
        if (p < NLEV) diff[(size_t)b * NLEV + p] = sv[r] - ST[p];
    }
}

// ---------------------------------------------------------------------------
// Phase 2: D[p] = sum_b diff[b,p] via WMMA f32 16x16x4 with A = ones.
// C[m,n] = sum_k B[k,n]; with A constant-1 the result is invariant to the
// K-slot placement inside the B operand. One wave per 16-column tile.
// Column index is clamped (not predicated) so every load is in-bounds and
// branch-free; out-of-range columns are zeroed with a select. Four
// independent accumulators break the WMMA RAW chain (4-way ILP).
// ---------------------------------------------------------------------------
__global__ __launch_bounds__(32)
void colsum_wmma_kernel(const float* __restrict__ diff, float* __restrict__ Dv)
{
    int lane = threadIdx.x & 31;
    int n    = lane & 15;
    int kb   = (lane >> 4) << 1;          // K pair: lanes 0-15 -> {0,1}, 16-31 -> {2,3}
    int col  = blockIdx.x * 16 + n;
    bool inC = col < NLEV;
    int colc = inC ? col : (NLEV - 1);    // clamped -> all addresses in-bounds

    v8f acc[4];
#pragma unroll
    for (int u = 0; u < 4; ++u) acc[u] = (v8f){0.f,0.f,0.f,0.f,0.f,0.f,0.f,0.f};
    v2f a = {1.f, 1.f};                   // all-ones A matrix (16x4)

    for (int k = 0; k < BATCH_N; k += 16) {
#pragma unroll
        for (int u = 0; u < 4; ++u) {
            const float* r = diff + (size_t)(k + 4 * u + kb) * NLEV + colc;
            float v0 = r[0];
            float v1 = r[NLEV];
            v2f bv;
            bv.x = inC ? v0 : 0.f;
            bv.y = inC ? v1 : 0.f;
            acc[u] = __builtin_amdgcn_wmma_f32_16x16x4_f32(
                         /*neg_a=*/false, a, /*neg_b=*/false, bv,
                         /*c_mod=*/(short)0, acc[u],
                         /*reuse_a=*/false, /*reuse_b=*/false);
        }
    }

    // every row of C holds column sums; row M=0 lives in element 0, lanes 0-15
    if ((lane < 16) && inC) {
        Dv[col] = ((acc[0][0] + acc[1][0]) + (acc[2][0] + acc[3][0]));
    }
}

// ---------------------------------------------------------------------------
// Phase 3: out = sum_p D[p]^2 / B^2 (fixed-order tree reduction)
// ---------------------------------------------------------------------------
__global__ __launch_bounds__(256)
void finalize_kernel(const float* __restrict__ Dv, float* __restrict__ out)
{
    __shared__ float red[256];
    int tid = threadIdx.x;
    float s = 0.f;
    for (int p = tid; p < NLEV; p += 256) {
        float d = Dv[p];
        s += d * d;
    }
    red[tid] = s;
    __syncthreads();
    for (int off = 128; off > 0; off >>= 1) {
        if (tid < off) red[tid] += red[tid + off];
        __syncthreads();
    }
    if (tid == 0) out[0] = red[0] * (1.f / ((float)BATCH_N * (float)BATCH_N));
}

// ---------------------------------------------------------------------------
extern "C" void kernel_launch(void* const* d_in, const int* in_sizes, int n_in,
                              void* d_out, int out_size, void* d_ws, size_t ws_size,
                              hipStream_t stream)
{
    (void)in_sizes; (void)n_in; (void)out_size; (void)ws_size;

    const float* x     = (const float*)d_in[0];
    const float* y     = (const float*)d_in[1];
    const float* sigma = (const float*)d_in[2];

    float* diff = (float*)d_ws;                          // 2048 * 780 floats
    float* Dv   = diff + (size_t)BATCH_N * NLEV;         // 780 floats
    float* out  = (float*)d_out;

    sigdiff_kernel<<<BATCH_N / WAVES, WAVES * 32, 0, stream>>>(x, y, sigma, diff);
    colsum_wmma_kernel<<<(NLEV + 15) / 16, 32, 0, stream>>>(diff, Dv);
    finalize_kernel<<<1, 256, 0, stream>>>(Dv, out);
}